// GuidedAnchorHeadWraper_1202590843777
// MI455X (gfx1250) — compile-verified
//
#include <hip/hip_runtime.h>
#include <hip/hip_bf16.h>
#include <math.h>

// ---------------- problem constants ----------------
#define BATCH    8
#define NCLS     80
#define NA       20460          // 15360+3840+960+240+60
#define K_SEL    1000           // NMS_PRE == final top-k
#define MAXDET   100
#define SCORE_THR 0.05f
#define IOU_THR   0.5f
#define LOC_THR   0.01f
#define MRA 13.815510557964274f   // |log(1e-6)|
#define MRB 4.135166556742356f    // |log(16/1000)|
#define IMGW 1280.0f
#define IMGH 768.0f

typedef __attribute__((ext_vector_type(2))) float v2f;
typedef __attribute__((ext_vector_type(8))) float v8f;

struct InPtrs { const float* p[20]; };

__device__ __forceinline__ float sigm(float x) { return 1.0f / (1.0f + expf(-x)); }

__device__ __forceinline__ void level_of(int a, int& local, int& H, int& W, float& stride, int& lvl) {
    if (a < 15360)      { lvl = 0; local = a;         H = 96; W = 160; stride = 8.f;   }
    else if (a < 19200) { lvl = 1; local = a - 15360; H = 48; W = 80;  stride = 16.f;  }
    else if (a < 20160) { lvl = 2; local = a - 19200; H = 24; W = 40;  stride = 32.f;  }
    else if (a < 20400) { lvl = 3; local = a - 20160; H = 12; W = 20;  stride = 64.f;  }
    else                { lvl = 4; local = a - 20400; H = 6;  W = 10;  stride = 128.f; }
}

// ------------------------------------------------------------------
// Kernel 1: per-anchor key (= mask * dsig(max logit)) + proposal box
// ------------------------------------------------------------------
__global__ void k_percell(InPtrs in, float* __restrict__ key, float* __restrict__ prop) {
    int idx = blockIdx.x * blockDim.x + threadIdx.x;
    if (idx >= BATCH * NA) return;
    int b = idx / NA, a = idx - b * NA;
    int local, H, W, lvl; float stride;
    level_of(a, local, H, W, stride, lvl);
    int hw = H * W;

    const float* cls = in.p[4 * lvl + 0];
    const float* bbx = in.p[4 * lvl + 1];
    const float* shp = in.p[4 * lvl + 2];
    const float* loc = in.p[4 * lvl + 3];

    // max over 80 classes (coalesced across lanes; stride hw per class)
    float m = -3.4e38f;
    int base = b * NCLS * hw + local;
    #pragma unroll 8
    for (int c = 0; c < NCLS; ++c) m = fmaxf(m, cls[base + c * hw]);

    float sl   = sigm(loc[b * hw + local]);
    float mask = (sl >= LOC_THR) ? 1.0f : 0.0f;
    key[idx] = mask * sigm(sigm(m));   // all nonzero keys are in (0.5, 0.7311)

    // guided anchor: square center (w*s, h*s), base side 4*s, grow by shape pred
    int h = local / W, w = local - h * W;
    float cx = w * stride, cy = h * stride;
    float bs = 4.0f * stride;
    float dwa = fminf(fmaxf(shp[b * 2 * hw + local],      -MRA), MRA);
    float dha = fminf(fmaxf(shp[b * 2 * hw + hw + local], -MRA), MRA);
    float aw = bs * expf(dwa), ah = bs * expf(dha);

    // bbox decode on the guided anchor, clip to image
    float dx  = bbx[b * 4 * hw + local];
    float dy  = bbx[b * 4 * hw + hw + local];
    float dbw = fminf(fmaxf(bbx[b * 4 * hw + 2 * hw + local], -MRB), MRB);
    float dbh = fminf(fmaxf(bbx[b * 4 * hw + 3 * hw + local], -MRB), MRB);
    float gx = cx + aw * dx, gy = cy + ah * dy;
    float gw = aw * expf(dbw), gh = ah * expf(dbh);
    float x1 = fminf(fmaxf(gx - 0.5f * gw, 0.f), IMGW);
    float y1 = fminf(fmaxf(gy - 0.5f * gh, 0.f), IMGH);
    float x2 = fminf(fmaxf(gx + 0.5f * gw, 0.f), IMGW);
    float y2 = fminf(fmaxf(gy + 0.5f * gh, 0.f), IMGH);
    prop[idx * 4 + 0] = x1; prop[idx * 4 + 1] = y1;
    prop[idx * 4 + 2] = x2; prop[idx * 4 + 3] = y2;
}

// ------------------------------------------------------------------
// Kernel 2: exact top-1000 per image via 4-pass byte radix-select.
// 256-bin suffix sums computed with V_WMMA_F32_16X16X4_F32:
//   rowSuffix = H(16x16) x U, U[k][n] = (k>=n), 4 chained K=4 WMMAs.
// f32 WMMA keeps counts (<=20460) exact.
// ------------------------------------------------------------------
__global__ void k_select(const float* __restrict__ key, int* __restrict__ selIdx) {
    int b = blockIdx.x;
    const float* kb = key + b * NA;
    __shared__ unsigned hist[256];
    __shared__ float histf[256];
    __shared__ float rs[256];        // row-wise suffix (WMMA result)
    __shared__ float tail[16];       // sum of totals of higher rows
    __shared__ unsigned suff[256];   // full 256-bin suffix counts
    __shared__ unsigned s_prefix, s_mask, s_kneed;
    __shared__ int s_cntG, s_cntT;
    int tid = threadIdx.x;

    if (tid == 0) { s_prefix = 0u; s_mask = 0u; s_kneed = K_SEL; }
    __syncthreads();

    for (int p = 3; p >= 0; --p) {
        int shift = p * 8;
        if (tid < 256) hist[tid] = 0u;
        __syncthreads();
        unsigned pref = s_prefix, msk = s_mask;
        for (int j = tid; j < NA; j += 1024) {
            __builtin_prefetch(kb + j + 2048, 0, 1);     // global_prefetch_b8
            unsigned u = __float_as_uint(kb[j]);
            if ((u & msk) == pref) atomicAdd(&hist[(u >> shift) & 255u], 1u);
        }
        __syncthreads();
        if (tid < 256) histf[tid] = (float)hist[tid];
        __syncthreads();

        // wave 0 (EXEC all ones): rowSuffix[r][n] = sum_{k>=n} H[r][k]
        if (tid < 32) {
#if __has_builtin(__builtin_amdgcn_wmma_f32_16x16x4_f32)
            int m = tid & 15, kh = tid >> 4;     // A: lanes0-15 K{0,1}, lanes16-31 K{2,3}
            v8f acc = {0.f, 0.f, 0.f, 0.f, 0.f, 0.f, 0.f, 0.f};
            #pragma unroll
            for (int kk = 0; kk < 4; ++kk) {
                int k0 = kk * 4 + 2 * kh;
                v2f av, bv;
                av.x = histf[m * 16 + k0];
                av.y = histf[m * 16 + k0 + 1];
                bv.x = (k0     >= m) ? 1.0f : 0.0f;  // B column n = lane&15 = m
                bv.y = (k0 + 1 >= m) ? 1.0f : 0.0f;
                acc = __builtin_amdgcn_wmma_f32_16x16x4_f32(
                    false, av, false, bv, (short)0, acc, false, false);
            }
            #pragma unroll
            for (int j = 0; j < 8; ++j) rs[(j + 8 * kh) * 16 + m] = acc[j];
#else
            for (int d = tid; d < 256; d += 32) {
                float s = 0.f;
                int r = d >> 4;
                for (int c = d & 15; c < 16; ++c) s += histf[r * 16 + c];
                rs[d] = s;
            }
#endif
        }
        __syncthreads();
        if (tid == 0) {
            float t = 0.f;
            for (int r = 15; r >= 0; --r) { tail[r] = t; t += rs[r * 16]; }
        }
        __syncthreads();
        if (tid < 256) suff[tid] = (unsigned)(rs[tid] + tail[tid >> 4] + 0.5f);
        __syncthreads();
        if (tid == 0) {
            unsigned kn = s_kneed;
            int d = 255;                       // largest d with suff[d] >= kn
            while (d > 0 && suff[d] < kn) --d;
            unsigned below = (d == 255) ? 0u : suff[d + 1];
            s_kneed = kn - below;
            s_prefix |= ((unsigned)d) << shift;
            s_mask   |= 0xFFu << shift;
        }
        __syncthreads();
    }

    if (tid == 0) { s_cntG = 0; s_cntT = 0; }
    __syncthreads();
    unsigned T = s_prefix;
    int needT = (int)s_kneed;
    int G = K_SEL - needT;                    // count of keys strictly > T
    for (int j = tid; j < NA; j += 1024) {
        unsigned u = __float_as_uint(kb[j]);
        if (u > T) {
            int pos = atomicAdd(&s_cntG, 1);
            selIdx[b * K_SEL + pos] = j;
        } else if (u == T) {
            int t = atomicAdd(&s_cntT, 1);
            if (t < needT) selIdx[b * K_SEL + G + t] = j;
        }
    }
}

// ------------------------------------------------------------------
// Kernel 3: gather 80-class thresholded scores + boxes for selection
// ------------------------------------------------------------------
__global__ void k_gather(InPtrs in, const float* __restrict__ key,
                         const float* __restrict__ prop,
                         const int* __restrict__ selIdx,
                         float* __restrict__ sv, float* __restrict__ selBoxes) {
    int idx = blockIdx.x * blockDim.x + threadIdx.x;
    if (idx >= BATCH * K_SEL * NCLS) return;
    int b = idx / (K_SEL * NCLS);
    int r = idx - b * K_SEL * NCLS;
    int k = r / NCLS, c = r - k * NCLS;
    int a = selIdx[b * K_SEL + k];
    float kv = key[b * NA + a];
    float s = 0.f;
    if (kv > 0.f) {            // mask==1 iff key>0 (dsig always > 0.5)
        int local, H, W, lvl; float stride;
        level_of(a, local, H, W, stride, lvl);
        int hw = H * W;
        float logit = in.p[4 * lvl][b * NCLS * hw + c * hw + local];
        float sc = sigm(sigm(logit));
        s = (sc > SCORE_THR) ? sc : 0.f;
    }
    sv[idx] = s;                                  // [b][k*80+c]
    if (c < 4) selBoxes[(b * K_SEL + k) * 4 + c] = prop[(b * NA + a) * 4 + c];
}

// ------------------------------------------------------------------
// Kernel 4: greedy class-aware NMS, one workgroup per image.
// Class offset 4096 > image width => cross-class IoU == 0, so
// suppression only touches the picked class (1000 IoUs / step).
// ------------------------------------------------------------------
__global__ void k_nms(float* __restrict__ sv, const float* __restrict__ selBoxes,
                      float* __restrict__ dout) {
    int b = blockIdx.x;
    int tid = threadIdx.x;
    float* svb = sv + b * (K_SEL * NCLS);
    __shared__ float lb[K_SEL * 4];
    __shared__ float rv[1024];
    __shared__ int   ri[1024];
    __shared__ int s_i, s_keep;

    for (int j = tid; j < K_SEL * 4; j += 1024) lb[j] = selBoxes[b * K_SEL * 4 + j];
    __syncthreads();

    float* outNum = dout;                              // [8]
    float* outBox = dout + 8;                          // [8][100][4]
    float* outSc  = dout + 8 + BATCH * MAXDET * 4;     // [8][100]
    float* outCl  = outSc + BATCH * MAXDET;            // [8][100]

    int cnt = 0;
    for (int t = 0; t < MAXDET; ++t) {
        // block-wide argmax with first-index tie-break
        float bv = -1.f; int bi = 0;
        for (int j = tid; j < K_SEL * NCLS; j += 1024) {
            float v = svb[j];
            if (v > bv) { bv = v; bi = j; }
        }
        rv[tid] = bv; ri[tid] = bi;
        __syncthreads();
        for (int s = 512; s > 0; s >>= 1) {
            if (tid < s) {
                if (rv[tid + s] > rv[tid] ||
                    (rv[tid + s] == rv[tid] && ri[tid + s] < ri[tid])) {
                    rv[tid] = rv[tid + s];
                    ri[tid] = ri[tid + s];
                }
            }
            __syncthreads();
        }
        if (tid == 0) {
            float sc = rv[0]; int i = ri[0];
            int keep = (sc > 0.f) ? 1 : 0;
            s_i = i; s_keep = keep;
            int k0 = i / NCLS, ci = i - k0 * NCLS;
            float kf = keep ? 1.f : 0.f;
            outBox[(b * MAXDET + t) * 4 + 0] = lb[k0 * 4 + 0] * kf;
            outBox[(b * MAXDET + t) * 4 + 1] = lb[k0 * 4 + 1] * kf;
            outBox[(b * MAXDET + t) * 4 + 2] = lb[k0 * 4 + 2] * kf;
            outBox[(b * MAXDET + t) * 4 + 3] = lb[k0 * 4 + 3] * kf;
            outSc[b * MAXDET + t] = sc * kf;
            outCl[b * MAXDET + t] = keep ? (float)ci : -1.f;
            cnt += keep;
        }
        __syncthreads();
        if (s_keep) {   // block-uniform
            int i = s_i, k0 = i / NCLS, ci = i - k0 * NCLS;
            float ax1 = lb[k0 * 4 + 0], ay1 = lb[k0 * 4 + 1];
            float ax2 = lb[k0 * 4 + 2], ay2 = lb[k0 * 4 + 3];
            float a1 = (ax2 - ax1) * (ay2 - ay1);
            for (int k = tid; k < K_SEL; k += 1024) {
                float bx1 = lb[k * 4 + 0], by1 = lb[k * 4 + 1];
                float bx2 = lb[k * 4 + 2], by2 = lb[k * 4 + 3];
                float iw = fmaxf(fminf(ax2, bx2) - fmaxf(ax1, bx1), 0.f);
                float ih = fmaxf(fminf(ay2, by2) - fmaxf(ay1, by1), 0.f);
                float inter = iw * ih;
                float a2 = (bx2 - bx1) * (by2 - by1);
                float iou = inter / (a1 + a2 - inter + 1e-6f);
                if (iou > IOU_THR) svb[k * NCLS + ci] = 0.f;
            }
        }
        __syncthreads();
    }
    if (tid == 0) outNum[b] = (float)cnt;
}

// ------------------------------------------------------------------
extern "C" void kernel_launch(void* const* d_in, const int* in_sizes, int n_in,
                              void* d_out, int out_size, void* d_ws, size_t ws_size,
                              hipStream_t stream) {
    (void)in_sizes; (void)n_in; (void)out_size; (void)ws_size;
    InPtrs ptrs;
    for (int i = 0; i < 20; ++i) ptrs.p[i] = (const float*)d_in[i];

    char* ws = (char*)d_ws;
    float* key      = (float*)(ws);                                   // 654,720 B
    float* prop     = (float*)(ws + 654720);                          // 2,618,880 B
    int*   selIdx   = (int*)  (ws + 654720 + 2618880);                // 32,000 B
    float* selBoxes = (float*)(ws + 654720 + 2618880 + 32000);        // 128,000 B
    float* sv       = (float*)(ws + 654720 + 2618880 + 32000 + 128000); // 2,560,000 B

    int n1 = BATCH * NA;
    k_percell<<<(n1 + 255) / 256, 256, 0, stream>>>(ptrs, key, prop);
    k_select<<<BATCH, 1024, 0, stream>>>(key, selIdx);
    int n3 = BATCH * K_SEL * NCLS;
    k_gather<<<(n3 + 255) / 256, 256, 0, stream>>>(ptrs, key, prop, selIdx, sv, selBoxes);
    k_nms<<<BATCH, 1024, 0, stream>>>(sv, selBoxes, (float*)d_out);
}